// MultiheadSelfAttention_21577915695370
// MI455X (gfx1250) — compile-verified
//
#include <hip/hip_runtime.h>
#include <hip/hip_bf16.h>

typedef __attribute__((ext_vector_type(2))) float v2f;
typedef __attribute__((ext_vector_type(8))) float v8f;

#define kB 8
#define kS 1025
#define kN 1024
#define kE 768
#define kH 12
#define kD 64

static __device__ __forceinline__ v8f wmma4(v2f a, v2f b, v8f c) {
  // D(16x16,f32) = A(16x4,f32) x B(4x16,f32) + C
  return __builtin_amdgcn_wmma_f32_16x16x4_f32(false, a, false, b, (short)0, c,
                                               false, false);
}

// ---------------------------------------------------------------------------
// Generic WMMA f32 GEMM:  Y = X @ W^T + bias     (W is N x K row-major)
// inMode: 0 = plain rows (row m -> X row m)
//         1 = skip-cls   (row m -> batch b=m/nTok, token i=m%nTok, X row
//                         b*(nTok+1) + 1 + i)         [QKV projections]
//         2 = cls-only   (row m -> X row m*nTok)      [cls projection]
// outMode: 0 = row-major M x 768
//          1 = head layout (B,H,nTok,D): col = h*64+d
// ---------------------------------------------------------------------------
__global__ __launch_bounds__(256) void wmma_gemm_kernel(
    const float* __restrict__ X, const float* __restrict__ W,
    const float* __restrict__ bias, float* __restrict__ Y, int M, int K,
    int inMode, int nTok, int outMode) {
  __shared__ float As[64][33];
  __shared__ float Bs[64][33];

  const int tid = threadIdx.x;
  const int lane = tid & 31;
  const int w = tid >> 5;
  const int m0 = blockIdx.x * 64;
  const int n0 = blockIdx.y * 64;
  const int wm = (w & 3) * 16;   // wave's 16 rows within tile
  const int wn = (w >> 2) * 32;  // wave's 32 cols within tile

  v8f acc0 = {};
  v8f acc1 = {};

  const int am = lane & 15;
  const int kpair = (lane >> 4) * 2;

  for (int k0 = 0; k0 < K; k0 += 32) {
    for (int idx = tid; idx < 64 * 32; idx += 256) {
      int r = idx >> 5, c = idx & 31;
      int m = m0 + r;
      float val = 0.f;
      if (m < M) {
        size_t xrow;
        if (inMode == 0) {
          xrow = (size_t)m;
        } else if (inMode == 1) {
          int b = m / nTok, i = m - b * nTok;
          xrow = (size_t)b * (nTok + 1) + 1 + i;
        } else {
          xrow = (size_t)m * nTok;
        }
        val = X[xrow * K + k0 + c];
      }
      As[r][c] = val;
    }
    for (int idx = tid; idx < 64 * 32; idx += 256) {
      int r = idx >> 5, c = idx & 31;
      Bs[r][c] = W[(size_t)(n0 + r) * K + k0 + c];
    }
    __syncthreads();

#pragma unroll
    for (int kk = 0; kk < 32; kk += 4) {
      v2f a;
      a.x = As[wm + am][kk + kpair];
      a.y = As[wm + am][kk + kpair + 1];
      v2f b0;
      b0.x = Bs[wn + am][kk + kpair];
      b0.y = Bs[wn + am][kk + kpair + 1];
      v2f b1;
      b1.x = Bs[wn + 16 + am][kk + kpair];
      b1.y = Bs[wn + 16 + am][kk + kpair + 1];
      acc0 = wmma4(a, b0, acc0);
      acc1 = wmma4(a, b1, acc1);
    }
    __syncthreads();
  }

  const int mrow_base = wm + (lane >> 4) * 8;
  const int ncol0 = n0 + wn + am;
  const int ncol1 = ncol0 + 16;
#pragma unroll
  for (int r = 0; r < 8; ++r) {
    int m = m0 + mrow_base + r;
    if (m >= M) continue;
    float y0 = acc0[r] + bias[ncol0];
    float y1 = acc1[r] + bias[ncol1];
    if (outMode == 0) {
      Y[(size_t)m * kE + ncol0] = y0;
      Y[(size_t)m * kE + ncol1] = y1;
    } else {
      int b = m / nTok, i = m - b * nTok;
      int h0 = ncol0 >> 6, d0 = ncol0 & 63;
      int h1 = ncol1 >> 6, d1 = ncol1 & 63;
      Y[((size_t)(b * kH + h0) * nTok + i) * kD + d0] = y0;
      Y[((size_t)(b * kH + h1) * nTok + i) * kD + d1] = y1;
    }
  }
}

// ---------------------------------------------------------------------------
// Per-row stats: q_norm2, c.q, c.k  (one thread per (b,h,token))
// ---------------------------------------------------------------------------
__global__ void row_stats_kernel(const float* __restrict__ q,
                                 const float* __restrict__ k,
                                 const float* __restrict__ c,
                                 float* __restrict__ qn2,
                                 float* __restrict__ cq,
                                 float* __restrict__ ck) {
  int idx = blockIdx.x * blockDim.x + threadIdx.x;
  if (idx >= kB * kH * kN) return;
  int bh = idx >> 10;
  int b = bh / kH, h = bh % kH;
  const float* crow = c + (size_t)b * kE + h * kD;
  const float* qrow = q + (size_t)idx * kD;
  const float* krow = k + (size_t)idx * kD;
  float s_qq = 0.f, s_cq = 0.f, s_ck = 0.f;
#pragma unroll 4
  for (int d = 0; d < kD; ++d) {
    float qd = qrow[d], kd = krow[d], cd = crow[d];
    s_qq += qd * qd;
    s_cq += cd * qd;
    s_ck += cd * kd;
  }
  qn2[idx] = s_qq;
  cq[idx] = s_cq;
  ck[idx] = s_ck;
}

__global__ void cn2_kernel(const float* __restrict__ c,
                           float* __restrict__ cn2) {
  int bh = blockIdx.x * blockDim.x + threadIdx.x;
  if (bh >= kB * kH) return;
  int b = bh / kH, h = bh % kH;
  const float* crow = c + (size_t)b * kE + h * kD;
  float s = 0.f;
  for (int d = 0; d < kD; ++d) s += crow[d] * crow[d];
  cn2[bh] = s;
}

// ---------------------------------------------------------------------------
// Fused attention: per block = one (b,h) and 16 query rows.
//  - logits via WMMA (Q tile in LDS, K read through cache)
//  - per-row scale cq/(qn2*sqrt(D*cn2)), softmax (cross-wave via LDS)
//  - attn written to d_out; O = P@V via per-wave LDS restaging + ds_add_f32
// ---------------------------------------------------------------------------
__global__ __launch_bounds__(256) void attn_kernel(
    const float* __restrict__ q, const float* __restrict__ k,
    const float* __restrict__ v, const float* __restrict__ qn2,
    const float* __restrict__ cq, const float* __restrict__ cn2,
    float* __restrict__ attn_out, float* __restrict__ vals_out) {
  const int bh = blockIdx.y;
  const int b = bh / kH, h = bh % kH;
  const int q0 = blockIdx.x * 16;
  const int tid = threadIdx.x;
  const int lane = tid & 31;
  const int w = tid >> 5;
  const int am = lane & 15;
  const int hi = lane >> 4;
  const int kpair = hi * 2;

  const float* qp = q + (size_t)bh * kN * kD;
  const float* kp = k + (size_t)bh * kN * kD;
  const float* vp = v + (size_t)bh * kN * kD;

  __shared__ float Qs[16][68];
  __shared__ float redbuf[8][16];
  __shared__ float stage[8][16][17];
  __shared__ float Osum[16][64];

  for (int idx = tid; idx < 16 * 64; idx += 256) {
    int r = idx >> 6, c = idx & 63;
    Qs[r][c] = qp[(size_t)(q0 + r) * kD + c];
    Osum[r][c] = 0.f;
  }
  __syncthreads();

  const float cn = fmaxf(cn2[bh], 1e-5f);
  const float inv_sq = 1.0f / sqrtf((float)kD * cn);
  float rscale[8];
#pragma unroll
  for (int r = 0; r < 8; ++r) {
    int row = q0 + hi * 8 + r;
    float qn = fmaxf(qn2[(size_t)bh * kN + row], 1e-5f);
    rscale[r] = cq[(size_t)bh * kN + row] * inv_sq / qn;
  }

  // ---- logits: wave w owns key columns [w*128, w*128+128) ----
  v8f sacc[8];
#pragma unroll
  for (int t = 0; t < 8; ++t) {
    v8f acc = {};
    const int c0 = w * 128 + t * 16;
    const float* kcol = kp + (size_t)(c0 + am) * kD;
#pragma unroll
    for (int kk = 0; kk < kD; kk += 4) {
      v2f a;
      a.x = Qs[am][kk + kpair];
      a.y = Qs[am][kk + kpair + 1];
      v2f bf;
      bf.x = kcol[kk + kpair];
      bf.y = kcol[kk + kpair + 1];
      acc = wmma4(a, bf, acc);
    }
#pragma unroll
    for (int r = 0; r < 8; ++r) acc[r] *= rscale[r];
    sacc[t] = acc;
  }

  // ---- row max (intra-wave shuffle, cross-wave via LDS) ----
  float rmax[8];
#pragma unroll
  for (int r = 0; r < 8; ++r) rmax[r] = -3.4e38f;
#pragma unroll
  for (int t = 0; t < 8; ++t)
#pragma unroll
    for (int r = 0; r < 8; ++r) rmax[r] = fmaxf(rmax[r], sacc[t][r]);
#pragma unroll
  for (int off = 1; off < 16; off <<= 1)
#pragma unroll
    for (int r = 0; r < 8; ++r)
      rmax[r] = fmaxf(rmax[r], __shfl_xor(rmax[r], off, 32));
  if (am == 0) {
#pragma unroll
    for (int r = 0; r < 8; ++r) redbuf[w][hi * 8 + r] = rmax[r];
  }
  __syncthreads();
  float gmax[8];
#pragma unroll
  for (int r = 0; r < 8; ++r) {
    float m = redbuf[0][hi * 8 + r];
#pragma unroll
    for (int ww = 1; ww < 8; ++ww) m = fmaxf(m, redbuf[ww][hi * 8 + r]);
    gmax[r] = m;
  }
  __syncthreads();

  // ---- exp + row sum ----
  float rsum[8];
#pragma unroll
  for (int r = 0; r < 8; ++r) rsum[r] = 0.f;
#pragma unroll
  for (int t = 0; t < 8; ++t)
#pragma unroll
    for (int r = 0; r < 8; ++r) {
      float p = expf(sacc[t][r] - gmax[r]);
      sacc[t][r] = p;
      rsum[r] += p;
    }
#pragma unroll
  for (int off = 1; off < 16; off <<= 1)
#pragma unroll
    for (int r = 0; r < 8; ++r) rsum[r] += __shfl_xor(rsum[r], off, 32);
  if (am == 0) {
#pragma unroll
    for (int r = 0; r < 8; ++r) redbuf[w][hi * 8 + r] = rsum[r];
  }
  __syncthreads();
  float invsum[8];
#pragma unroll
  for (int r = 0; r < 8; ++r) {
    float s = 0.f;
#pragma unroll
    for (int ww = 0; ww < 8; ++ww) s += redbuf[ww][hi * 8 + r];
    invsum[r] = 1.0f / s;
  }

  // ---- write normalized attn to global output ----
  float* arow = attn_out + (size_t)bh * kN * kN;
#pragma unroll
  for (int t = 0; t < 8; ++t) {
    int col = w * 128 + t * 16 + am;
#pragma unroll
    for (int r = 0; r < 8; ++r) {
      int row = q0 + hi * 8 + r;
      arow[(size_t)row * kN + col] = sacc[t][r] * invsum[r];
    }
  }

  // ---- O partial = P_w (16x128) @ V[w*128:+128, :] via WMMA ----
  v8f oacc[4] = {};
#pragma unroll
  for (int t = 0; t < 8; ++t) {
    __syncthreads();
#pragma unroll
    for (int r = 0; r < 8; ++r) stage[w][hi * 8 + r][am] = sacc[t][r];
    __syncthreads();
    const int kbase = w * 128 + t * 16;
#pragma unroll
    for (int kk = 0; kk < 16; kk += 4) {
      v2f a;
      a.x = stage[w][am][kk + kpair];
      a.y = stage[w][am][kk + kpair + 1];
#pragma unroll
      for (int dt = 0; dt < 4; ++dt) {
        v2f bf;
        bf.x = vp[(size_t)(kbase + kk + kpair) * kD + dt * 16 + am];
        bf.y = vp[(size_t)(kbase + kk + kpair + 1) * kD + dt * 16 + am];
        oacc[dt] = wmma4(a, bf, oacc[dt]);
      }
    }
  }

#pragma unroll
  for (int dt = 0; dt < 4; ++dt)
#pragma unroll
    for (int r = 0; r < 8; ++r)
      atomicAdd(&Osum[hi * 8 + r][dt * 16 + am], oacc[dt][r] * invsum[r]);
  __syncthreads();

  float* vrow = vals_out + ((size_t)b * kS + 1 + q0) * kE + h * kD;
  for (int idx = tid; idx < 16 * 64; idx += 256) {
    int r = idx >> 6, c = idx & 63;
    vrow[(size_t)r * kE + c] = Osum[r][c];
  }
}

// ---------------------------------------------------------------------------
// cls path: cls_out = (cls_token + (ck/sqrt(D*cn2)) @ v) * 0.5  -> vals row 0
// ---------------------------------------------------------------------------
__global__ void cls_kernel(const float* __restrict__ x,
                           const float* __restrict__ v,
                           const float* __restrict__ ck,
                           const float* __restrict__ cn2,
                           float* __restrict__ vals_out) {
  int idx = blockIdx.x * blockDim.x + threadIdx.x;
  if (idx >= kB * kH * kD) return;
  int d = idx & 63;
  int bh = idx >> 6;
  int b = bh / kH, h = bh % kH;
  float cn = fmaxf(cn2[bh], 1e-5f);
  float sc = 1.0f / sqrtf((float)kD * cn);
  const float* vp = v + (size_t)bh * kN * kD;
  const float* ckp = ck + (size_t)bh * kN;
  float acc = 0.f;
  for (int kk = 0; kk < kN; ++kk) acc += ckp[kk] * vp[(size_t)kk * kD + d];
  acc *= sc;
  float clstok = x[(size_t)b * kS * kE + h * kD + d];
  vals_out[(size_t)b * kS * kE + h * kD + d] = (clstok + acc) * 0.5f;
}

// ---------------------------------------------------------------------------
extern "C" void kernel_launch(void* const* d_in, const int* in_sizes, int n_in,
                              void* d_out, int out_size, void* d_ws,
                              size_t ws_size, hipStream_t stream) {
  (void)in_sizes; (void)n_in; (void)out_size; (void)ws_size;
  const float* x = (const float*)d_in[0];
  const float* Wq = (const float*)d_in[1];
  const float* bq = (const float*)d_in[2];
  const float* Wk = (const float*)d_in[3];
  const float* bk = (const float*)d_in[4];
  const float* Wv = (const float*)d_in[5];
  const float* bv = (const float*)d_in[6];
  const float* Wc = (const float*)d_in[7];
  const float* bc = (const float*)d_in[8];
  const float* Wo = (const float*)d_in[9];
  const float* bo = (const float*)d_in[10];

  float* ws = (float*)d_ws;
  const size_t qkvSz = (size_t)kB * kH * kN * kD;  // 6291456 floats
  float* qb = ws;
  float* kb = qb + qkvSz;
  float* vb = kb + qkvSz;
  float* cb = vb + qkvSz;                 // (B, E)
  float* qn2 = cb + (size_t)kB * kE;      // (B*H*n)
  float* cqb = qn2 + (size_t)kB * kH * kN;
  float* ckb = cqb + (size_t)kB * kH * kN;
  float* cn2 = ckb + (size_t)kB * kH * kN;  // (B*H)
  float* vals = cn2 + 128;                  // (B, S, E)

  float* outp = (float*)d_out;
  float* attnp = outp + (size_t)kB * kS * kE;

  dim3 blk(256);

  // Q, K, V projections (head layout outputs)
  wmma_gemm_kernel<<<dim3(kB * kN / 64, kE / 64), blk, 0, stream>>>(
      x, Wq, bq, qb, kB * kN, kE, 1, kN, 1);
  wmma_gemm_kernel<<<dim3(kB * kN / 64, kE / 64), blk, 0, stream>>>(
      x, Wk, bk, kb, kB * kN, kE, 1, kN, 1);
  wmma_gemm_kernel<<<dim3(kB * kN / 64, kE / 64), blk, 0, stream>>>(
      x, Wv, bv, vb, kB * kN, kE, 1, kN, 1);
  // cls projection (M=8)
  wmma_gemm_kernel<<<dim3(1, kE / 64), blk, 0, stream>>>(
      x, Wc, bc, cb, kB, kE, 2, kS, 0);

  cn2_kernel<<<1, 128, 0, stream>>>(cb, cn2);
  row_stats_kernel<<<(kB * kH * kN) / 256, blk, 0, stream>>>(qb, kb, cb, qn2,
                                                             cqb, ckb);

  attn_kernel<<<dim3(kN / 16, kB * kH), blk, 0, stream>>>(
      qb, kb, vb, qn2, cqb, cn2, attnp, vals);

  cls_kernel<<<(kB * kH * kD) / 256, blk, 0, stream>>>(x, vb, ckb, cn2, vals);

  // out = vals @ Wo^T + bo   (M = 8200, guarded)
  wmma_gemm_kernel<<<dim3((kB * kS + 63) / 64, kE / 64), blk, 0, stream>>>(
      vals, Wo, bo, outp, kB * kS, kE, 0, 1, 0);
}